// BackprojectDepth_corre_18253611008840
// MI455X (gfx1250) — compile-verified
//
#include <hip/hip_runtime.h>

// Problem constants (from the reference).
#define BATCH   16
#define IMG_H   384
#define IMG_W   1280
#define IMG_HW  (IMG_H * IMG_W)       // 491520
#define KSEL    131072
#define TILES_PER_B   (KSEL / 16)     // 8192 16-column WMMA tiles per batch
#define TOTAL_TILES   (BATCH * TILES_PER_B)  // 131072
#define NUM_BLOCKS    2048
#define THREADS       256             // 8 waves per block (wave32)
#define NUM_WAVES     (NUM_BLOCKS * THREADS / 32)  // 16384
#define TILES_PER_WAVE (TOTAL_TILES / NUM_WAVES)   // 8 (all in one b: 8192 % 8 == 0)

typedef __attribute__((ext_vector_type(2))) float v2f;
typedef __attribute__((ext_vector_type(8))) float v8f;

// D(16x16,f32) = A(16x4,f32) x B(4x16,f32) + 0
// A operand layout (wave32, 2 VGPRs): lane<16 : {K=0, K=1}; lane>=16 : {K=2, K=3}, row M = lane&15
// B operand layout (by symmetry / C-D striping): lane<16 : rows K=0,K=1 at col N=lane;
//                                                lane>=16: rows K=2,K=3 at col N=lane-16
// D layout: VGPR r, lanes 0-15 -> M=r, N=lane (rows 0..2 are what we need).
__global__ __launch_bounds__(THREADS) void backproject_wmma_kernel(
    const float* __restrict__ depth,   // (B, 1, H, W) flat
    const float* __restrict__ invK,    // (B, 4, 4) flat
    const int*   __restrict__ topk,    // (B, K) flat
    float*       __restrict__ out)     // (B, 4, K) flat
{
    const int lane       = threadIdx.x & 31;
    const int waveGlobal = blockIdx.x * (THREADS >> 5) + (threadIdx.x >> 5);
    const int t0         = waveGlobal * TILES_PER_WAVE;
    const int b          = t0 / TILES_PER_B;
    const int k0         = (t0 - b * TILES_PER_B) * 16;

    // ---- Build A operand once per wave: inv_K[b][0:3][0:3] padded into 16x4 ----
    const float* iK = invK + b * 16;
    v2f amat; amat.x = 0.0f; amat.y = 0.0f;
    if (lane < 3) {                       // rows 0..2, K=0 / K=1
        amat.x = iK[lane * 4 + 0];
        amat.y = iK[lane * 4 + 1];
    } else if (lane >= 16 && lane < 19) { // rows 0..2, K=2 (K=3 stays 0)
        amat.x = iK[(lane - 16) * 4 + 2];
    }

    const int  col   = lane & 15;
    const bool lower = (lane < 16);
    const int*   tkb = topk  + b * KSEL;
    const float* dpb = depth + (size_t)b * IMG_HW;
    float*       ob  = out   + (size_t)b * 4 * KSEL;

    for (int i = 0; i < TILES_PER_WAVE; ++i) {
        const int kk = k0 + i * 16 + col;

        int   idx  = 0;
        float dsel = 1.0f;
        if (lower) {
            idx  = tkb[kk];     // streamed, coalesced
            dsel = dpb[idx];    // random gather; batch slice (1.9 MB) lives in L2
            if (i + 1 < TILES_PER_WAVE)
                __builtin_prefetch(&tkb[kk + 16], 0, 1);  // global_prefetch_b8
        }

        // pix_coords is the meshgrid: recompute (gx,gy,1) from idx instead of
        // gathering 3 floats from the 94 MB pix_coords tensor.
        const int gy = (int)((unsigned)idx / IMG_W);
        const int gx = idx - gy * IMG_W;

        v2f bmat;
        bmat.x = lower ? (float)gx : 1.0f;  // lanes>=16: K=2 row == ones
        bmat.y = lower ? (float)gy : 0.0f;  // lanes>=16: K=3 row == zeros

        v8f cacc = {};
        // D = A x B + 0  (v_wmma_f32_16x16x4_f32), EXEC is full here.
        v8f d = __builtin_amdgcn_wmma_f32_16x16x4_f32(
            /*neg_a=*/false, amat, /*neg_b=*/false, bmat,
            /*c_mod=*/(short)0, cacc, /*reuse_a=*/false, /*reuse_b=*/false);

        if (lower) {
            float* o = ob + kk;
            o[0 * KSEL] = dsel * d[0];  // x * depth
            o[1 * KSEL] = dsel * d[1];  // y * depth
            o[2 * KSEL] = dsel * d[2];  // z * depth
            o[3 * KSEL] = 1.0f;         // ones row
        }
    }
}

extern "C" void kernel_launch(void* const* d_in, const int* in_sizes, int n_in,
                              void* d_out, int out_size, void* d_ws, size_t ws_size,
                              hipStream_t stream) {
    const float* depth = (const float*)d_in[0];  // (B,1,H,W) f32
    const float* invK  = (const float*)d_in[1];  // (B,4,4)   f32
    // d_in[2] = pix_coords — intentionally unused (analytically recomputed)
    const int*   topk  = (const int*)d_in[3];    // (B,K)     i32
    float*       out   = (float*)d_out;          // (B,4,K)   f32

    hipLaunchKernelGGL(backproject_wmma_kernel,
                       dim3(NUM_BLOCKS), dim3(THREADS), 0, stream,
                       depth, invK, topk, out);
}